// get_loss_4758823764601
// MI455X (gfx1250) — compile-verified
//
#include <hip/hip_runtime.h>
#include <hip/hip_bf16.h>

typedef float v2f __attribute__((ext_vector_type(2)));
typedef float v8f __attribute__((ext_vector_type(8)));

// ---------------------------------------------------------------------------
// Prep 1: augmented B matrix (target side), one float4 per target point m:
//   Baug[m] = { tx, ty, tz, ||t||^2 }
// ---------------------------------------------------------------------------
__global__ void prep_target_kernel(const float* __restrict__ target,
                                   float4* __restrict__ Baug, int M) {
    int m = blockIdx.x * blockDim.x + threadIdx.x;
    if (m < M) {
        float x = target[3 * m + 0];
        float y = target[3 * m + 1];
        float z = target[3 * m + 2];
        Baug[m] = make_float4(x, y, z, x * x + y * y + z * z);
    }
}

// ---------------------------------------------------------------------------
// Prep 2: augmented A matrix (source - trans_m[s]), one float4 per (s,n) row:
//   Aaug[s*N+n] = { -2*px, -2*py, -2*pz, 1 },  a2[s*N+n] = ||p||^2
// A(row) . B(col) = ||b||^2 - 2 a.b ; add ||a||^2 after min/max.
// ---------------------------------------------------------------------------
__global__ void prep_source_kernel(const float* __restrict__ source,
                                   const float* __restrict__ trans,
                                   float4* __restrict__ Aaug,
                                   float* __restrict__ a2, int N, int SN) {
    int j = blockIdx.x * blockDim.x + threadIdx.x;
    if (j < SN) {
        int n = j % N;
        float x = source[3 * n + 0] - trans[3 * j + 0];
        float y = source[3 * n + 1] - trans[3 * j + 1];
        float z = source[3 * n + 2] - trans[3 * j + 2];
        Aaug[j] = make_float4(-2.f * x, -2.f * y, -2.f * z, 1.0f);
        a2[j]   = x * x + y * y + z * z;
    }
}

// ---------------------------------------------------------------------------
// Main kernel: one wave per 16-row A tile (rows in the combined S*N space).
// Inner loop over M/16 column tiles: load B frag (b64), v_wmma_f32_16x16x4_f32
// with C=0, elementwise running min/max over the 8 C-layout VGPRs.
//
// Fragment layouts per ISA 7.12.2 (wave32):
//  A 16x4 f32 : lane L -> row base+(L&15), VGPR j -> K = j + 2*(L>=16)
//  B 4x16 f32 : lane L -> col tile*(16)+(L&15), VGPR j -> K = j + 2*(L>=16)
//  C 16x16 f32: VGPR v: lanes 0-15 row v, lanes 16-31 row v+8, col = L&15
// ---------------------------------------------------------------------------
__global__ void chamfer_wmma_kernel(const float4* __restrict__ Aaug,
                                    const float4* __restrict__ Baug,
                                    const float* __restrict__ a2,
                                    const int* __restrict__ all_stage_p,
                                    float* __restrict__ partial,
                                    int N, int M, int S) {
    const int lane = threadIdx.x & 31;
    const int wid  = (blockIdx.x * blockDim.x + threadIdx.x) >> 5;
    const int tilesPerStage = N >> 4;
    const int totalTiles    = S * tilesPerStage;
    if (wid >= totalTiles) return;          // wave-uniform; EXEC stays all-1s

    const int rowBase = wid << 4;
    const int half    = lane >> 4;          // 0 or 1
    const int sub     = lane & 15;

    // A fragment is loop-invariant: float2 slice of the augmented row.
    const v2f* ap = (const v2f*)((const float*)(Aaug + (rowBase + sub)) + 2 * half);
    const v2f afrag = *ap;

    v8f vmin, vmax;
#pragma unroll
    for (int i = 0; i < 8; ++i) {
        vmin[i] = __builtin_inff();
        vmax[i] = -__builtin_inff();
    }

    const v8f czero = {0.f, 0.f, 0.f, 0.f, 0.f, 0.f, 0.f, 0.f};
    const int colTiles = M >> 4;

#pragma unroll 4
    for (int ct = 0; ct < colTiles; ++ct) {
        const v2f* bp =
            (const v2f*)((const float*)(Baug + (ct * 16 + sub)) + 2 * half);
        const v2f bfrag = *bp;

        // D = A x B  (16x16 tile of  ||b||^2 - 2 a.b)
        v8f d = __builtin_amdgcn_wmma_f32_16x16x4_f32(
            /*neg_a=*/false, afrag, /*neg_b=*/false, bfrag,
            /*c_mod=*/(short)0, czero, /*reuse_a=*/false, /*reuse_b=*/false);

#pragma unroll
        for (int i = 0; i < 8; ++i) {
            vmin[i] = fminf(vmin[i], d[i]);
            vmax[i] = fmaxf(vmax[i], d[i]);
        }
    }

    // Column reduction: xor-shuffle masks 1,2,4,8 stay within 16-lane groups.
#pragma unroll
    for (int i = 0; i < 8; ++i) {
        float mn = vmin[i], mx = vmax[i];
#pragma unroll
        for (int msk = 1; msk < 16; msk <<= 1) {
            mn = fminf(mn, __shfl_xor(mn, msk, 32));
            mx = fmaxf(mx, __shfl_xor(mx, msk, 32));
        }
        vmin[i] = mn;
        vmax[i] = mx;
    }

    // Lane 0 owns rows rowBase+0..7, lane 16 owns rows rowBase+8..15.
    float sum = 0.f;
    if (sub == 0) {
#pragma unroll
        for (int i = 0; i < 8; ++i) {
            int row  = rowBase + 8 * half + i;
            float aa = a2[row];
            // 0.99*min_dist + 0.01*max_dist, dist = ||a||^2 + reduced value
            sum += 0.99f * (aa + vmin[i]) + 0.01f * (aa + vmax[i]);
        }
    }
    sum += __shfl_xor(sum, 16, 32);  // combine lanes 0 and 16

    if (lane == 0) {
        int   s = wid / tilesPerStage;
        float w = (s == S - 1) ? 1.0f : ((*all_stage_p == 1) ? 0.1f : 0.0f);
        partial[wid] = w * sum / (float)N;
    }
}

// ---------------------------------------------------------------------------
// Deterministic final reduction: fixed-order strided sums + LDS tree.
// Overwrites d_out[0] (handles workspace/out poisoning).
// ---------------------------------------------------------------------------
__global__ void final_reduce_kernel(const float* __restrict__ partial,
                                    float* __restrict__ out, int T) {
    __shared__ float sbuf[256];
    float s = 0.f;
    for (int i = threadIdx.x; i < T; i += 256) s += partial[i];
    sbuf[threadIdx.x] = s;
    __syncthreads();
    for (int stride = 128; stride > 0; stride >>= 1) {
        if ((int)threadIdx.x < stride) sbuf[threadIdx.x] += sbuf[threadIdx.x + stride];
        __syncthreads();
    }
    if (threadIdx.x == 0) out[0] = sbuf[0];
}

extern "C" void kernel_launch(void* const* d_in, const int* in_sizes, int n_in,
                              void* d_out, int out_size, void* d_ws, size_t ws_size,
                              hipStream_t stream) {
    const float* source    = (const float*)d_in[0];
    const float* target    = (const float*)d_in[1];
    const float* trans     = (const float*)d_in[2];
    const int*   all_stage = (const int*)d_in[5];

    const int N  = in_sizes[0] / 3;            // 8192
    const int M  = in_sizes[1] / 3;            // 8192
    const int S  = in_sizes[2] / (N * 3);      // 4
    const int SN = S * N;

    // Workspace layout (all 16B-aligned given aligned d_ws):
    //   Baug:    M   float4   (128 KB)
    //   Aaug:    S*N float4   (512 KB)
    //   a2:      S*N float    (128 KB)
    //   partial: S*N/16 float (  8 KB)
    char*   ws      = (char*)d_ws;
    float4* Baug    = (float4*)ws;
    float4* Aaug    = (float4*)(ws + (size_t)M * 16);
    float*  a2v     = (float*)(ws + (size_t)M * 16 + (size_t)SN * 16);
    float*  partial = (float*)(ws + (size_t)M * 16 + (size_t)SN * 16 +
                               (size_t)SN * 4);

    prep_target_kernel<<<(M + 255) / 256, 256, 0, stream>>>(target, Baug, M);
    prep_source_kernel<<<(SN + 255) / 256, 256, 0, stream>>>(source, trans,
                                                             Aaug, a2v, N, SN);

    const int totalTiles    = S * (N / 16);    // 2048 row tiles (one per wave)
    const int wavesPerBlock = 8;               // 256 threads = 8 wave32s
    const int blocks = (totalTiles + wavesPerBlock - 1) / wavesPerBlock;
    chamfer_wmma_kernel<<<blocks, 256, 0, stream>>>(Aaug, Baug, a2v, all_stage,
                                                    partial, N, M, S);

    final_reduce_kernel<<<1, 256, 0, stream>>>(partial, (float*)d_out,
                                               totalTiles);
}